// LabelMixReg_88553635709059
// MI455X (gfx1250) — compile-verified
//
#include <hip/hip_runtime.h>
#include <float.h>

// ---------------------------------------------------------------------------
// LabelMix regularization loss, MI455X (gfx1250, wave32).
// HBM-bound: 180 MB single-pass stream; 35-wide per-pixel seg-channel dot done on the
// WMMA pipe (V_WMMA_F32_16X16X4_F32, weights broadcast in A, pixels in B's
// N dimension) so argmax VALU work co-issues with the matrix pipe.
// ---------------------------------------------------------------------------

typedef __attribute__((ext_vector_type(2))) float v2f;
typedef __attribute__((ext_vector_type(8))) float v8f;

#define HWSZ   65536      // 256*256
#define NCH    35         // seg classes
#define CIMG   3
#define NPIX   (16 * HWSZ)        // 1,048,576 pixels
#define BLOCK  256
#define WPB    (BLOCK / 32)       // 8 waves per block
#define PPW    16                 // pixels per wave (N dim of WMMA)
#define NBLK   (NPIX / (WPB * PPW))   // 8192 blocks, exact cover

__global__ __launch_bounds__(BLOCK) void lmr_main_gfx1250(
    const float* __restrict__ real, const float* __restrict__ gen,
    const float* __restrict__ seg,  const float* __restrict__ rdisc,
    const float* __restrict__ gdisc,const float* __restrict__ w,
    const int*   __restrict__ lut,  float* __restrict__ partials)
{
    __shared__ float smem[WPB];

    const int tid  = threadIdx.x;
    const int lane = tid & 31;
    const int wv   = tid >> 5;
    const int half = lane >> 4;     // 0: K%4 in {0,1}; 1: K%4 in {2,3}
    const int pl   = lane & 15;     // pixel within wave tile (N index)

    const int gwave = blockIdx.x * WPB + wv;
    const int pix   = gwave * PPW + pl;          // tiles never cross batch
    const int b     = pix >> 16;
    const int hw    = pix & (HWSZ - 1);

    const float* segp = seg + (size_t)(b * NCH) * HWSZ + hw;

    // ---- fused pass: seg-channel argmax (VALU) + GEMV via WMMA (XDL) ----
    float mx = -FLT_MAX;
    int   mi = 0;
    v8f acc = {0.f, 0.f, 0.f, 0.f, 0.f, 0.f, 0.f, 0.f};

#pragma unroll
    for (int c = 0; c < 9; ++c) {               // K = 36 (ch 35 zero-padded)
        const int kA = 4 * c + 2 * half;        // this lane-half's K slots
        const int kB = kA + 1;
        const float sA = (kA < NCH) ? segp[(size_t)kA * HWSZ] : 0.f;
        const float sB = (kB < NCH) ? segp[(size_t)kB * HWSZ] : 0.f;
        const float wA = (kA < NCH) ? w[CIMG + kA] : 0.f;
        const float wB = (kB < NCH) ? w[CIMG + kB] : 0.f;

        // first-occurrence argmax (per-lane K strictly increasing)
        if (kA < NCH && sA > mx) { mx = sA; mi = kA; }
        if (kB < NCH && sB > mx) { mx = sB; mi = kB; }

        // A[m,k] = w[k] for all m (broadcast rows); B[k,n] = seg[k][pixel n]
        v2f Av = {wA, wB};
        v2f Bv = {sA, sB};
        acc = __builtin_amdgcn_wmma_f32_16x16x4_f32(
                  false, Av, false, Bv, (short)0, acc, false, false);
    }

    // D rows are identical (A rows identical) -> D vgpr0 lane L = dot(pixel L&15)
    const float segdot = acc[0];

    // merge the two K-halves' argmax (ties -> lower channel index, jnp semantics)
    const float mxo = __shfl_xor(mx, 16, 32);
    const int   mio = __shfl_xor(mi, 16, 32);
    if (mxo > mx || (mxo == mx && mio < mi)) { mx = mxo; mi = mio; }

    const float t = (float)lut[mi];
    const float u = 1.0f - t;

    // ---- image channels + disc target, split across lane halves ----
    const size_t ibase = (size_t)(b * CIMG) * HWSZ + hw;
    const size_t dbase = (size_t)b * HWSZ + hw;
    float part;
    if (half == 0) {
        const float m0 = t * real[ibase]                    + u * gen[ibase];
        const float m2 = t * real[ibase + 2 * (size_t)HWSZ] + u * gen[ibase + 2 * (size_t)HWSZ];
        part = m0 * w[0] + m2 * w[2] - t * rdisc[dbase];
    } else {
        const float m1 = t * real[ibase + (size_t)HWSZ] + u * gen[ibase + (size_t)HWSZ];
        part = m1 * w[1] - u * gdisc[dbase];
    }

    const float e  = segdot + part + __shfl_xor(part, 16, 32); // identical in both halves
    float sq = (half == 0) ? e * e : 0.f;                      // count each pixel once

    // ---- wave32 reduction (fixed order -> deterministic) ----
    sq += __shfl_xor(sq, 1, 32);
    sq += __shfl_xor(sq, 2, 32);
    sq += __shfl_xor(sq, 4, 32);
    sq += __shfl_xor(sq, 8, 32);
    sq += __shfl_xor(sq, 16, 32);
    if (lane == 0) smem[wv] = sq;
    __syncthreads();
    if (tid == 0) {
        float s = 0.f;
#pragma unroll
        for (int i = 0; i < WPB; ++i) s += smem[i];
        partials[blockIdx.x] = s;
    }
}

// deterministic fixed-order final reduction of the 8192 block partials
__global__ __launch_bounds__(256) void lmr_reduce_gfx1250(
    const float* __restrict__ partials, float* __restrict__ out)
{
    __shared__ float smem[256];
    float s = 0.f;
    for (int i = threadIdx.x; i < NBLK; i += 256) s += partials[i];
    smem[threadIdx.x] = s;
    __syncthreads();
    for (int off = 128; off > 0; off >>= 1) {
        if (threadIdx.x < off) smem[threadIdx.x] += smem[threadIdx.x + off];
        __syncthreads();
    }
    if (threadIdx.x == 0) out[0] = smem[0] * (1.0f / (float)NPIX);
}

extern "C" void kernel_launch(void* const* d_in, const int* in_sizes, int n_in,
                              void* d_out, int out_size, void* d_ws, size_t ws_size,
                              hipStream_t stream) {
    const float* real = (const float*)d_in[0];
    const float* gen  = (const float*)d_in[1];
    const float* seg  = (const float*)d_in[2];
    const float* rd   = (const float*)d_in[3];
    const float* gd   = (const float*)d_in[4];
    const float* w    = (const float*)d_in[5];
    const int*   lut  = (const int*)d_in[6];
    float* partials   = (float*)d_ws;   // 8192 floats = 32 KB scratch

    lmr_main_gfx1250<<<NBLK, BLOCK, 0, stream>>>(real, gen, seg, rd, gd, w, lut, partials);
    lmr_reduce_gfx1250<<<1, 256, 0, stream>>>(partials, (float*)d_out);
}